// VectorQuantizer_3736621548030
// MI455X (gfx1250) — compile-verified
//
#include <hip/hip_runtime.h>

typedef __attribute__((ext_vector_type(2))) float v2f;
typedef __attribute__((ext_vector_type(8))) float v8f;

#define NUM_CODES 1024
#define CODE_DIM  256
#define B_        32
#define T_        1024
#define NROWS     (B_ * T_)                 // 32768
#define QELEMS    (B_ * CODE_DIM * T_)      // 8388608

// ---------------------------------------------------------------------------
// Kernel 1: per-code squared norms  ww[k] = sum_d codebook[k][d]^2
// ---------------------------------------------------------------------------
__global__ __launch_bounds__(256) void vq_code_norms(const float* __restrict__ cb,
                                                     float* __restrict__ ww) {
    int k = blockIdx.x * blockDim.x + threadIdx.x;
    if (k >= NUM_CODES) return;
    const float4* row = (const float4*)(cb + (size_t)k * CODE_DIM);
    float s = 0.f;
#pragma unroll 8
    for (int i = 0; i < CODE_DIM / 4; ++i) {
        float4 v = row[i];
        s += v.x * v.x + v.y * v.y + v.z * v.z + v.w * v.w;
    }
    ww[k] = s;
}

// ---------------------------------------------------------------------------
// Kernel 2: fused GEMM (X . W^T via V_WMMA_F32_16X16X4_F32) + argmin.
// One workgroup (8 waves) per 32-row tile of X (two 16-row A tiles in LDS).
// 2x2 register blocking: each A fragment (LDS) and each B fragment (L2
// codebook) feeds TWO wmmas -> 4 wmmas per {2 ds_b64 + 2 global_b64},
// halving both LDS bandwidth per wmma and codebook L2 traffic.
// ---------------------------------------------------------------------------
__global__ __launch_bounds__(256) void vq_argmin(const float* __restrict__ latents,
                                                 const float* __restrict__ cb,
                                                 const float* __restrict__ ww,
                                                 float* __restrict__ out_idx_f,
                                                 int* __restrict__ ws_idx) {
    // Per tile: ldsA[tile*2048 + kk*16 + m] = (X[tile*16+m][2kk], X[tile*16+m][2kk+1])
    // A-fragment read for k = 4*s: ldsA[tile*2048 + s*32 + lane]   (ds_load_b64)
    __shared__ v2f   ldsA[4096];          // 32 KB (two 16x256 tiles)
    __shared__ float ldsWW[NUM_CODES];    // 4 KB
    __shared__ float redV[256];
    __shared__ int   redI[256];

    const int tid  = threadIdx.x;
    const int wave = tid >> 5;
    const int lane = tid & 31;
    const int col  = lane & 15;
    const int half = lane >> 4;

    const int row0 = blockIdx.x * 32;         // 32 rows, never crosses a b boundary
    const int b    = row0 >> 10;
    const int t0   = row0 & 1023;
    const float* lat = latents + (size_t)b * CODE_DIM * T_ + t0;

    // Stage both A tiles: X[tile*16+m][d] = latents[b][d][t0 + tile*16 + m]
#pragma unroll
    for (int i = 0; i < 16; ++i) {
        int e    = tid + i * 256;             // 0..4095
        int tile = e >> 11;
        int ee   = e & 2047;
        int kk   = ee >> 4;
        int m    = ee & 15;
        v2f a;
        a.x = lat[(size_t)(2 * kk    ) * T_ + tile * 16 + m];
        a.y = lat[(size_t)(2 * kk + 1) * T_ + tile * 16 + m];
        ldsA[e] = a;
    }
#pragma unroll
    for (int i = 0; i < 4; ++i) ldsWW[tid + i * 256] = ww[tid + i * 256];
    __syncthreads();

    const v8f vzero = {0.f, 0.f, 0.f, 0.f, 0.f, 0.f, 0.f, 0.f};
    float bestV[2][8];
    int   bestI[2][8];
#pragma unroll
    for (int t = 0; t < 2; ++t)
#pragma unroll
        for (int j = 0; j < 8; ++j) { bestV[t][j] = 3.4e38f; bestI[t][j] = 0x7fffffff; }

    const v2f* aptr0 = ldsA + lane;
    const v2f* aptr1 = ldsA + 2048 + lane;

    // wave owns N-tiles {wave + 8*q, q=0..7}; process them in pairs
    for (int p = 0; p < 4; ++p) {
        const int n0    = wave + 16 * p;
        const int n1    = n0 + 8;
        const int code0 = n0 * 16 + col;                   // this lane's code columns
        const int code1 = n1 * 16 + col;
        const float* bptr0 = cb + (size_t)code0 * CODE_DIM + 2 * half;
        const float* bptr1 = cb + (size_t)code1 * CODE_DIM + 2 * half;

        v8f acc00 = vzero;   // (tile0, n0)
        v8f acc10 = vzero;   // (tile1, n0)
        v8f acc01 = vzero;   // (tile0, n1)
        v8f acc11 = vzero;   // (tile1, n1)
#pragma unroll 4
        for (int s = 0; s < 64; ++s) {                     // k = 4*s
            v2f a0 = aptr0[s * 32];                        // ds_load_b64
            v2f a1 = aptr1[s * 32];                        // ds_load_b64
            v2f b0 = *(const v2f*)(bptr0 + s * 4);         // global_load_b64
            v2f b1 = *(const v2f*)(bptr1 + s * 4);         // global_load_b64
            acc00 = __builtin_amdgcn_wmma_f32_16x16x4_f32(
                false, a0, false, b0, (short)0, acc00, false, false);
            acc10 = __builtin_amdgcn_wmma_f32_16x16x4_f32(
                false, a1, false, b0, (short)0, acc10, false, false);
            acc01 = __builtin_amdgcn_wmma_f32_16x16x4_f32(
                false, a0, false, b1, (short)0, acc01, false, false);
            acc11 = __builtin_amdgcn_wmma_f32_16x16x4_f32(
                false, a1, false, b1, (short)0, acc11, false, false);
        }
        // dist(M, code) = ww[code] - 2*dot ; C layout: VGPR j -> M=j (+8 in hi half)
        const float ww0 = ldsWW[code0];
        const float ww1 = ldsWW[code1];
#pragma unroll
        for (int j = 0; j < 8; ++j) {
            float d;
            bool better;
            d = ww0 - 2.0f * acc00[j];
            better = (d < bestV[0][j]) | ((d == bestV[0][j]) & (code0 < bestI[0][j]));
            bestV[0][j] = better ? d : bestV[0][j];
            bestI[0][j] = better ? code0 : bestI[0][j];
            d = ww1 - 2.0f * acc01[j];
            better = (d < bestV[0][j]) | ((d == bestV[0][j]) & (code1 < bestI[0][j]));
            bestV[0][j] = better ? d : bestV[0][j];
            bestI[0][j] = better ? code1 : bestI[0][j];
            d = ww0 - 2.0f * acc10[j];
            better = (d < bestV[1][j]) | ((d == bestV[1][j]) & (code0 < bestI[1][j]));
            bestV[1][j] = better ? d : bestV[1][j];
            bestI[1][j] = better ? code0 : bestI[1][j];
            d = ww1 - 2.0f * acc11[j];
            better = (d < bestV[1][j]) | ((d == bestV[1][j]) & (code1 < bestI[1][j]));
            bestV[1][j] = better ? d : bestV[1][j];
            bestI[1][j] = better ? code1 : bestI[1][j];
        }
    }

    // reduce across the 16 lanes of each half (masks 1..8 stay within halves)
#pragma unroll
    for (int t = 0; t < 2; ++t) {
#pragma unroll
        for (int j = 0; j < 8; ++j) {
#pragma unroll
            for (int m = 1; m <= 8; m <<= 1) {
                float ov = __shfl_xor(bestV[t][j], m, 32);
                int   oi = __shfl_xor(bestI[t][j], m, 32);
                bool better = (ov < bestV[t][j]) | ((ov == bestV[t][j]) & (oi < bestI[t][j]));
                bestV[t][j] = better ? ov : bestV[t][j];
                bestI[t][j] = better ? oi : bestI[t][j];
            }
        }
    }
    if (col == 0) {                                        // lanes 0 and 16
#pragma unroll
        for (int t = 0; t < 2; ++t)
#pragma unroll
            for (int j = 0; j < 8; ++j) {
                int r = t * 16 + j + 8 * half;             // row within 32-row tile
                redV[wave * 32 + r] = bestV[t][j];
                redI[wave * 32 + r] = bestI[t][j];
            }
    }
    __syncthreads();

    // cross-wave reduce + write indices
    if (tid < 32) {
        float bv = redV[tid];
        int   bi = redI[tid];
#pragma unroll
        for (int w = 1; w < 8; ++w) {
            float v  = redV[w * 32 + tid];
            int   i2 = redI[w * 32 + tid];
            bool better = (v < bv) | ((v == bv) & (i2 < bi));
            bv = better ? v : bv;
            bi = better ? i2 : bi;
        }
        int row = row0 + tid;
        out_idx_f[row] = (float)bi;
        ws_idx[row]    = bi;
    }
}

// ---------------------------------------------------------------------------
// Kernel 3: gather quantized = codebook[idx] into [B, D, T] layout, and
// accumulate (q - x)^2 partial sums per block (deterministic in-block order).
// ---------------------------------------------------------------------------
__global__ __launch_bounds__(256) void vq_gather_loss(const float* __restrict__ latents,
                                                      const float* __restrict__ cb,
                                                      const int* __restrict__ ws_idx,
                                                      float* __restrict__ outq,
                                                      float* __restrict__ partials) {
    __shared__ float red[256];
    const int tid = threadIdx.x;
    const int blk = blockIdx.x;            // 0..127
    const int b   = blk >> 2;
    const int t   = ((blk & 3) << 8) + tid;
    const int row = b * T_ + t;
    const int idx = ws_idx[row];

    const float4* q4   = (const float4*)(cb + (size_t)idx * CODE_DIM);
    const float*  latb = latents + (size_t)b * CODE_DIM * T_ + t;
    float*        outb = outq    + (size_t)b * CODE_DIM * T_ + t;

    float acc = 0.f;
#pragma unroll 4
    for (int d4 = 0; d4 < CODE_DIM / 4; ++d4) {
        float4 q = q4[d4];
        size_t base = (size_t)(d4 * 4) * T_;
        float x0 = latb[base + 0 * T_];
        float x1 = latb[base + 1 * T_];
        float x2 = latb[base + 2 * T_];
        float x3 = latb[base + 3 * T_];
        outb[base + 0 * T_] = q.x;
        outb[base + 1 * T_] = q.y;
        outb[base + 2 * T_] = q.z;
        outb[base + 3 * T_] = q.w;
        float e0 = q.x - x0, e1 = q.y - x1, e2 = q.z - x2, e3 = q.w - x3;
        acc += e0 * e0 + e1 * e1 + e2 * e2 + e3 * e3;
    }
    red[tid] = acc;
    __syncthreads();
    for (int s = 128; s > 0; s >>= 1) {
        if (tid < s) red[tid] += red[tid + s];
        __syncthreads();
    }
    if (tid == 0) partials[blk] = red[0];
}

// ---------------------------------------------------------------------------
// Kernel 4: deterministic final loss reduction. loss = 1.25 * mean((q-x)^2)
// ---------------------------------------------------------------------------
__global__ __launch_bounds__(128) void vq_loss_final(const float* __restrict__ partials,
                                                     float* __restrict__ out_loss) {
    __shared__ float red[128];
    const int tid = threadIdx.x;
    red[tid] = partials[tid];
    __syncthreads();
    for (int s = 64; s > 0; s >>= 1) {
        if (tid < s) red[tid] += red[tid + s];
        __syncthreads();
    }
    if (tid == 0) {
        float mse = red[0] / (float)QELEMS;
        out_loss[0] = mse * 1.25f;   // codebook_loss + 0.25 * commitment_loss (equal)
    }
}

// ---------------------------------------------------------------------------
extern "C" void kernel_launch(void* const* d_in, const int* in_sizes, int n_in,
                              void* d_out, int out_size, void* d_ws, size_t ws_size,
                              hipStream_t stream) {
    const float* latents = (const float*)d_in[0];   // [32, 256, 1024]
    const float* cb      = (const float*)d_in[1];   // [1024, 256]

    float* out      = (float*)d_out;
    float* outq     = out;                          // [QELEMS]
    float* out_loss = out + QELEMS;                 // [1]
    float* out_idx  = out + QELEMS + 1;             // [NROWS] (as float)

    float* ws_ww   = (float*)d_ws;                               // 1024 f32
    int*   ws_idx  = (int*)  ((char*)d_ws + 4096);               // 32768 i32
    float* ws_part = (float*)((char*)d_ws + 4096 + 131072);      // 128 f32

    vq_code_norms<<<NUM_CODES / 256, 256, 0, stream>>>(cb, ws_ww);
    vq_argmin<<<NROWS / 32, 256, 0, stream>>>(latents, cb, ws_ww, out_idx, ws_idx);
    vq_gather_loss<<<128, 256, 0, stream>>>(latents, cb, ws_idx, outq, ws_part);
    vq_loss_final<<<1, 128, 0, stream>>>(ws_part, out_loss);
}